// S4Layer_62362925138077
// MI455X (gfx1250) — compile-verified
//
#include <hip/hip_runtime.h>

// S4 layer discretization step for MI455X (gfx1250).
//
// Math: A is lower triangular (HiPPO), so (I - (d/2)A)^-1 applied to
// [(I + (d/2)A) x + d B u] is: one batched GEMM  M = X * A^T  (WMMA f32 path,
// exact fp32), then a per-head forward substitution (one lane per head).
//
// One wave (32 threads) per workgroup handles 32 heads. 256 workgroups.

#define H_TOTAL 8192
#define NSTATE 64
#define HPB 32           // heads per block
#define STRIDE 68        // padded LDS row stride (floats): conflict-free WMMA operand fetch

typedef float v8f __attribute__((ext_vector_type(8)));
typedef float v2f __attribute__((ext_vector_type(2)));

__global__ __launch_bounds__(32) void s4_step_kernel(
    const float* __restrict__ x_state,   // (H, 64)
    const float* __restrict__ u_input,   // (H,)
    const float* __restrict__ A,         // (64, 64) lower triangular
    const float* __restrict__ Bvec,      // (64, 1)
    const float* __restrict__ C,         // (H, 64)
    const float* __restrict__ log_delta, // (H,)
    float* __restrict__ out_state,       // (H, 64)
    float* __restrict__ out_y)           // (H,)
{
    __shared__ float lds_A[NSTATE * STRIDE];   // A, padded rows (row-major)
    __shared__ float lds_x[HPB * STRIDE];      // x tile, later reused for C tile
    __shared__ float lds_v[HPB * STRIDE];      // M = X*A^T, then RHS v
    __shared__ float lds_s[HPB * STRIDE];      // solution s
    __shared__ float lds_B[NSTATE];

    const int t    = threadIdx.x;              // 0..31
    const int base = blockIdx.x * HPB;         // first head of this block

    // ---- Stage A (64x64) into padded LDS rows, coalesced float4 ----
    for (int it = 0; it < 32; ++it) {
        int p4  = (t + 32 * it) * 4;           // covers 4096 floats
        int row = p4 >> 6, col = p4 & 63;
        float4 d = *(const float4*)(A + p4);
        *(float4*)(lds_A + row * STRIDE + col) = d;
    }
    // ---- Stage X tile (32x64) ----
    for (int it = 0; it < 16; ++it) {
        int p4  = (t + 32 * it) * 4;           // covers 2048 floats
        int row = p4 >> 6, col = p4 & 63;
        float4 d = *(const float4*)(x_state + base * NSTATE + p4);
        *(float4*)(lds_x + row * STRIDE + col) = d;
    }
    lds_B[t]      = Bvec[t];
    lds_B[t + 32] = Bvec[t + 32];
    __syncthreads();

    // ---- GEMM: M = X_tile * A^T via V_WMMA_F32_16X16X4_F32 ----
    // A-operand (16x4 f32): lane l<16 -> row M=l, {K0,K1}; lane>=16 -> M=l-16, {K2,K3}.
    // B-operand (4x16 f32, assumed mirror): lane l<16 -> col N=l, rows {K0,K1};
    //                                        lane>=16 -> N=l-16, rows {K2,K3}.
    // B[K][N] = A^T[k][n] = A[16*tt + N][4*kk + K]  -> contiguous float2 in lds_A.
    const int lhalf = t & 15;
    const int hi    = t >> 4;                  // 0 or 1
    for (int g = 0; g < 2; ++g) {              // head group: heads g*16 .. g*16+15
        for (int tt = 0; tt < 4; ++tt) {       // output column tile
            v8f c = {0.f, 0.f, 0.f, 0.f, 0.f, 0.f, 0.f, 0.f};
            #pragma unroll
            for (int kk = 0; kk < 16; ++kk) {  // K = 64 in steps of 4
                int koff = 4 * kk + 2 * hi;
                v2f a = *(const v2f*)(lds_x + (g * 16 + lhalf) * STRIDE + koff);
                v2f b = *(const v2f*)(lds_A + (16 * tt + lhalf) * STRIDE + koff);
                c = __builtin_amdgcn_wmma_f32_16x16x4_f32(
                        /*neg_a=*/false, a, /*neg_b=*/false, b,
                        /*c_mod=*/(short)0, c, /*reuse_a=*/false, /*reuse_b=*/false);
            }
            // D layout: VGPR r = (head r, lanes 0-15) / (head r+8, lanes 16-31), N = lane%16
            #pragma unroll
            for (int r = 0; r < 8; ++r) {
                int head = g * 16 + r + 8 * hi;
                lds_v[head * STRIDE + 16 * tt + lhalf] = c[r];
            }
        }
    }
    __syncthreads();

    // ---- Per-lane head setup + RHS: v = x + (d/2)*M + d*B*u ----
    const int   hl    = t;
    const int   h     = base + hl;
    const float delta = __expf(log_delta[h]);
    const float hd    = 0.5f * delta;
    const float du    = delta * u_input[h];
    #pragma unroll 4
    for (int i = 0; i < NSTATE; ++i) {
        lds_v[hl * STRIDE + i] =
            lds_x[hl * STRIDE + i] + hd * lds_v[hl * STRIDE + i] + du * lds_B[i];
    }
    __syncthreads();

    // ---- Reuse lds_x for the C tile ----
    for (int it = 0; it < 16; ++it) {
        int p4  = (t + 32 * it) * 4;
        int row = p4 >> 6, col = p4 & 63;
        float4 d = *(const float4*)(C + base * NSTATE + p4);
        *(float4*)(lds_x + row * STRIDE + col) = d;
    }
    __syncthreads();

    // ---- Forward substitution: (I - (d/2)A) s = v ; A lower triangular ----
    // s_i = (v_i + (d/2) * sum_{j<i} A_ij s_j) / (1 + (d/2)*(i+1))
    // lds_A reads are same-address across lanes (broadcast); s reads are own-row.
    for (int i = 0; i < NSTATE; ++i) {
        float sum = 0.f;
        for (int j = 0; j < i; ++j)
            sum += lds_A[i * STRIDE + j] * lds_s[hl * STRIDE + j];
        float si = (lds_v[hl * STRIDE + i] + hd * sum) /
                   (1.0f + hd * (float)(i + 1));
        lds_s[hl * STRIDE + i] = si;
    }

    // ---- y = C . s (own-row reads only, no sync needed yet) ----
    float acc = 0.f;
    #pragma unroll 4
    for (int i = 0; i < NSTATE; ++i)
        acc += lds_x[hl * STRIDE + i] * lds_s[hl * STRIDE + i];
    out_y[h] = acc;

    __syncthreads();  // cross-head reads of lds_s below

    // ---- Coalesced float4 write-back of new_state ----
    for (int it = 0; it < 16; ++it) {
        int p4  = (t + 32 * it) * 4;
        int row = p4 >> 6, col = p4 & 63;
        float4 d = *(const float4*)(lds_s + row * STRIDE + col);
        *(float4*)(out_state + base * NSTATE + p4) = d;
    }
}

extern "C" void kernel_launch(void* const* d_in, const int* in_sizes, int n_in,
                              void* d_out, int out_size, void* d_ws, size_t ws_size,
                              hipStream_t stream) {
    const float* x_state   = (const float*)d_in[0];  // (8192, 64)
    const float* u_input   = (const float*)d_in[1];  // (8192,)
    const float* A         = (const float*)d_in[2];  // (64, 64)
    const float* Bvec      = (const float*)d_in[3];  // (64, 1)
    const float* C         = (const float*)d_in[4];  // (8192, 64)
    const float* log_delta = (const float*)d_in[5];  // (8192,)

    float* out_state = (float*)d_out;                    // first H*N floats
    float* out_y     = out_state + H_TOTAL * NSTATE;     // next H floats

    dim3 grid(H_TOTAL / HPB);   // 256 workgroups
    dim3 block(HPB);            // 32 threads = 1 wave32
    s4_step_kernel<<<grid, block, 0, stream>>>(x_state, u_input, A, Bvec, C,
                                               log_delta, out_state, out_y);
}